// FreeYOLO_5781025981019
// MI455X (gfx1250) — compile-verified
//
#include <hip/hip_runtime.h>
#include <stdint.h>

// ---------------- problem constants ----------------
#define NCLS   80
#define TOPK   1000
#define NLVL   3
#define NDET   (NLVL * TOPK)     // 3000
#define NPAD   3008              // 188 * 16
#define NB     8192              // histogram bins over score in [0,1]
#define WMSK   96                // mask words per row (3072 bits >= NPAD)
#define NSRT   4096              // bitonic sort size
#define CAP    4096              // eq-bin candidate capacity per level
#define IMGF   2560.0f
#define CONF_T 0.05f
#define NMS_THR 0.6f

typedef __attribute__((ext_vector_type(2))) float v2f;
typedef __attribute__((ext_vector_type(8))) float v8f;

// ---------------- workspace layout (bytes) ----------------
#define OFF_HIST      0                         // 3*NB*4
#define OFF_META      (OFF_HIST + NLVL*NB*4)    // 64 ints
#define OFF_CAND      (OFF_META + 256)          // 3*CAP*8 u64 keys
#define OFF_DET_SCORE (OFF_CAND + NLVL*CAP*8)
#define OFF_DET_LABEL (OFF_DET_SCORE + NDET*4)
#define OFF_DET_VALID (OFF_DET_LABEL + NDET*4)
#define OFF_DET_BOX   (OFF_DET_VALID + NDET*4)
#define OFF_S_SCORE   (OFF_DET_BOX + NDET*16)
#define OFF_S_LABEL   (OFF_S_SCORE + NPAD*4)
#define OFF_S_VALID   (OFF_S_LABEL + NPAD*4)
#define OFF_S_BOX     (OFF_S_VALID + NPAD*4)
#define OFF_S_BOFS    (OFF_S_BOX + NPAD*16)
#define OFF_S_AREA    (OFF_S_BOFS + NPAD*16)
#define OFF_MASK      (OFF_S_AREA + NPAD*4)
#define WS_END        (OFF_MASK + NDET*WMSK*4)

// META int layout: [0..2]=thresh_bin  [3..5]=need_eq  [6..8]=eq_cnt  [9..11]=hi_cnt

__device__ __forceinline__ float sigm(float x) { return 1.0f / (1.0f + __expf(-x)); }
__device__ __forceinline__ float fuse_score(float o, float c) { return sqrtf(sigm(o) * sigm(c)); }
__device__ __forceinline__ int bin_of(float s) {
    int b = (int)(s * (float)NB);
    return b < 0 ? 0 : (b > NB - 1 ? NB - 1 : b);
}

__device__ __forceinline__ void decode_det(int g, int t, float s,
                                           const float* __restrict__ reg,
                                           const float* __restrict__ anc, float stride,
                                           float* det_score, int* det_label, int* det_valid,
                                           float* det_box) {
    int m = t / NCLS;
    int lbl = t - m * NCLS;
    float rx = reg[m * 4 + 0], ry = reg[m * 4 + 1];
    float rw = reg[m * 4 + 2], rh = reg[m * 4 + 3];
    float cx = anc[m * 2 + 0] + rx * stride;
    float cy = anc[m * 2 + 1] + ry * stride;
    float w = __expf(rw) * stride, h = __expf(rh) * stride;
    det_box[g * 4 + 0] = cx - 0.5f * w;
    det_box[g * 4 + 1] = cy - 0.5f * h;
    det_box[g * 4 + 2] = cx + 0.5f * w;
    det_box[g * 4 + 3] = cy + 0.5f * h;
    det_score[g] = s;
    det_label[g] = lbl;
    det_valid[g] = (s > CONF_T) ? 1 : 0;
}

// ---------------- K0: zero workspace ----------------
__global__ void k_zero(unsigned* __restrict__ w, int n) {
    for (int i = blockIdx.x * blockDim.x + threadIdx.x; i < n; i += gridDim.x * blockDim.x)
        w[i] = 0u;
}

// ---------------- K1: streaming score histogram ----------------
__global__ void __launch_bounds__(256) k_score_hist(const float* __restrict__ obj,
                                                    const float* __restrict__ cls, int total,
                                                    unsigned* __restrict__ hist) {
    __shared__ unsigned lh[NB];
    for (int i = threadIdx.x; i < NB; i += blockDim.x) lh[i] = 0u;
    __syncthreads();
    int stride = gridDim.x * blockDim.x;
    for (int t = blockIdx.x * blockDim.x + threadIdx.x; t < total; t += stride) {
        __builtin_prefetch(cls + t + 4 * stride, 0, 1);  // -> global_prefetch_b8
        int m = t / NCLS;
        float s = fuse_score(obj[m], cls[t]);
        atomicAdd(&lh[bin_of(s)], 1u);
    }
    __syncthreads();
    for (int i = threadIdx.x; i < NB; i += blockDim.x)
        if (lh[i]) atomicAdd(&hist[i], lh[i]);
}

// ---------------- K2: find per-level threshold bin ----------------
__global__ void k_thresh(const unsigned* __restrict__ hist, int* __restrict__ meta) {
    int l = threadIdx.x;
    if (l < NLVL) {
        const unsigned* h = hist + l * NB;
        unsigned cum = 0;
        int b = NB - 1;
        for (; b >= 0; --b) {
            unsigned nc = cum + h[b];
            if (nc >= TOPK) break;
            cum = nc;
        }
        if (b < 0) b = 0;
        meta[0 + l] = b;            // threshold bin
        meta[3 + l] = TOPK - (int)cum;  // take this many from bin b
    }
}

// ---------------- K3a: compact (hi-bin -> direct decode, eq-bin -> candidates) ------
__global__ void __launch_bounds__(256) k_compact(const float* __restrict__ obj,
                                                 const float* __restrict__ cls, int total, int lvl,
                                                 int* __restrict__ meta,
                                                 unsigned long long* __restrict__ cand,
                                                 const float* __restrict__ reg,
                                                 const float* __restrict__ anc, float stride,
                                                 float* det_score, int* det_label, int* det_valid,
                                                 float* det_box) {
    int tb = meta[0 + lvl];
    int gs = gridDim.x * blockDim.x;
    for (int t = blockIdx.x * blockDim.x + threadIdx.x; t < total; t += gs) {
        __builtin_prefetch(cls + t + 4 * gs, 0, 1);
        int m = t / NCLS;
        float s = fuse_score(obj[m], cls[t]);
        int b = bin_of(s);
        if (b > tb) {
            int slot = atomicAdd(&meta[9 + lvl], 1);
            if (slot < TOPK)
                decode_det(lvl * TOPK + slot, t, s, reg, anc, stride, det_score, det_label,
                           det_valid, det_box);
        } else if (b == tb) {
            int p = atomicAdd(&meta[6 + lvl], 1);
            if (p < CAP) {
                // composite key: score bits (desc) then smaller index first
                unsigned long long k =
                    ((unsigned long long)__float_as_uint(s) << 32) | (unsigned)(~(unsigned)t);
                cand[lvl * CAP + p] = k;
            }
        }
    }
}

// ---------------- K3b: deterministic selection from the threshold bin ----------------
__global__ void __launch_bounds__(256) k_eq_select(int* __restrict__ meta,
                                                   unsigned long long* __restrict__ cand_all,
                                                   const float* r0, const float* r1, const float* r2,
                                                   const float* a0, const float* a1, const float* a2,
                                                   float* det_score, int* det_label, int* det_valid,
                                                   float* det_box) {
    int l = blockIdx.x;
    const float* reg = (l == 0) ? r0 : (l == 1) ? r1 : r2;
    const float* anc = (l == 0) ? a0 : (l == 1) ? a1 : a2;
    float stride = (l == 0) ? 8.0f : (l == 1) ? 16.0f : 32.0f;
    const unsigned long long* cand = cand_all + l * CAP;

    __shared__ unsigned long long key[NSRT];  // 32 KB
    int tid = threadIdx.x, bd = blockDim.x;
    int cnt = meta[6 + l];
    if (cnt > CAP) cnt = CAP;
    for (int i = tid; i < NSRT; i += bd) key[i] = (i < cnt) ? cand[i] : 0ULL;
    __syncthreads();
    // bitonic sort, descending
    for (unsigned kk = 2; kk <= NSRT; kk <<= 1) {
        for (unsigned j = kk >> 1; j > 0; j >>= 1) {
            for (unsigned i = tid; i < NSRT; i += bd) {
                unsigned ixj = i ^ j;
                if (ixj > i) {
                    unsigned long long A = key[i], B = key[ixj];
                    bool up = ((i & kk) == 0);  // descending segment
                    if (up ? (A < B) : (A > B)) { key[i] = B; key[ixj] = A; }
                }
            }
            __syncthreads();
        }
    }
    int hi = meta[9 + l];
    if (hi > TOPK) hi = TOPK;
    int need = meta[3 + l];
    if (need > TOPK - hi) need = TOPK - hi;
    if (need > cnt) need = cnt;
    if (need < 0) need = 0;
    for (int i = tid; i < need; i += bd) {
        unsigned long long k = key[i];
        float s = __uint_as_float((unsigned)(k >> 32));
        int t = (int)(~(unsigned)(k & 0xFFFFFFFFULL));
        decode_det(l * TOPK + hi + i, t, s, reg, anc, stride, det_score, det_label, det_valid,
                   det_box);
    }
}

// ---------------- K5: global sort (3000), offset boxes, areas ----------------
__global__ void __launch_bounds__(512) k_sort_prep(const float* __restrict__ det_score,
                                                   const int* __restrict__ det_label,
                                                   const int* __restrict__ det_valid,
                                                   const float* __restrict__ det_box,
                                                   float* s_score, int* s_label, int* s_valid,
                                                   float* s_box, float* s_bofs, float* s_area) {
    __shared__ unsigned long long key[NSRT];  // 32 KB
    __shared__ float red[512];
    __shared__ float sscale;
    int tid = threadIdx.x, bd = blockDim.x;
    for (int i = tid; i < NSRT; i += bd) {
        unsigned long long k = 0ULL;
        if (i < NDET) {
            unsigned hb = det_valid[i] ? __float_as_uint(det_score[i]) : 0u;
            k = ((unsigned long long)hb << 32) | (unsigned)(~(unsigned)i);
        }
        key[i] = k;
    }
    __syncthreads();
    for (unsigned kk = 2; kk <= NSRT; kk <<= 1) {
        for (unsigned j = kk >> 1; j > 0; j >>= 1) {
            for (unsigned i = tid; i < NSRT; i += bd) {
                unsigned ixj = i ^ j;
                if (ixj > i) {
                    unsigned long long A = key[i], B = key[ixj];
                    bool up = ((i & kk) == 0);
                    if (up ? (A < B) : (A > B)) { key[i] = B; key[ixj] = A; }
                }
            }
            __syncthreads();
        }
    }
    // max over all box coords -> class offset scale (matches jnp.max(boxes)+1)
    float mx = -3.0e38f;
    for (int i = tid; i < NDET * 4; i += bd) mx = fmaxf(mx, det_box[i]);
    red[tid] = mx;
    __syncthreads();
    for (int s = bd >> 1; s > 0; s >>= 1) {
        if (tid < s) red[tid] = fmaxf(red[tid], red[tid + s]);
        __syncthreads();
    }
    if (tid == 0) sscale = red[0] + 1.0f;
    __syncthreads();
    float scale = sscale;
    for (int i = tid; i < NPAD; i += bd) {
        if (i < NDET) {
            int src = (int)(~(unsigned)(key[i] & 0xFFFFFFFFULL));
            float sc = det_score[src];
            int lb = det_label[src];
            s_score[i] = sc;
            s_label[i] = lb;
            s_valid[i] = det_valid[src];
            float off = (float)lb * scale;
            float b0 = det_box[src * 4 + 0], b1 = det_box[src * 4 + 1];
            float b2 = det_box[src * 4 + 2], b3 = det_box[src * 4 + 3];
            s_box[i * 4 + 0] = b0; s_box[i * 4 + 1] = b1;
            s_box[i * 4 + 2] = b2; s_box[i * 4 + 3] = b3;
            s_bofs[i * 4 + 0] = b0 + off; s_bofs[i * 4 + 1] = b1 + off;
            s_bofs[i * 4 + 2] = b2 + off; s_bofs[i * 4 + 3] = b3 + off;
            s_area[i] = (b2 - b0) * (b3 - b1);
        } else {
            s_score[i] = 0.f; s_label[i] = 0; s_valid[i] = 0;
            s_box[i * 4 + 0] = 0.f; s_box[i * 4 + 1] = 0.f;
            s_box[i * 4 + 2] = 0.f; s_box[i * 4 + 3] = 0.f;
            s_bofs[i * 4 + 0] = 0.f; s_bofs[i * 4 + 1] = 0.f;
            s_bofs[i * 4 + 2] = 0.f; s_bofs[i * 4 + 3] = 0.f;
            s_area[i] = 0.f;
        }
    }
}

// ---------------- K6: 16x16 IoU tiles -> suppression bit-matrix (WMMA denominator) ----
// One wave per tile. area_i + area_j for the whole tile via V_WMMA_F32_16X16X4_F32:
//   A(16x4) rows = [area_i, 1, 0, 0], B(4x16) cols = [1, area_j, 0, 0]  =>  C_ij = area_i+area_j
__global__ void __launch_bounds__(32) k_mask(const float* __restrict__ bofs,
                                             const float* __restrict__ area,
                                             unsigned* __restrict__ mask) {
    int i0 = blockIdx.y * 16, j0 = blockIdx.x * 16;
    if (j0 + 15 <= i0) return;  // uniform: only tiles containing some j > i
    int lane = threadIdx.x;
    int half = lane >> 4;       // 0: lanes 0-15, 1: lanes 16-31
    int n = lane & 15;
    __shared__ float sbi[64], sbj[64], sai[16], saj[16];
    if (half == 0) {
        int r = i0 + n;
        sai[n] = area[r];
        sbi[n * 4 + 0] = bofs[r * 4 + 0]; sbi[n * 4 + 1] = bofs[r * 4 + 1];
        sbi[n * 4 + 2] = bofs[r * 4 + 2]; sbi[n * 4 + 3] = bofs[r * 4 + 3];
    } else {
        int cg = j0 + n;
        saj[n] = area[cg];
        sbj[n * 4 + 0] = bofs[cg * 4 + 0]; sbj[n * 4 + 1] = bofs[cg * 4 + 1];
        sbj[n * 4 + 2] = bofs[cg * 4 + 2]; sbj[n * 4 + 3] = bofs[cg * 4 + 3];
    }
    __syncthreads();

    // A 16x4 layout: lanes 0-15 hold K=0 (v0) / K=1 (v1); lanes 16-31 hold K=2/3 (zeros)
    v2f a, b;
    a.x = half ? 0.0f : sai[n];
    a.y = half ? 0.0f : 1.0f;
    b.x = half ? 0.0f : 1.0f;
    b.y = half ? 0.0f : saj[n];
    v8f acc = {};
    acc = __builtin_amdgcn_wmma_f32_16x16x4_f32(false, a, false, b, (short)0, acc, false, false);

    float bj0 = sbj[n * 4 + 0], bj1 = sbj[n * 4 + 1];
    float bj2 = sbj[n * 4 + 2], bj3 = sbj[n * 4 + 3];
    int j = j0 + n;
    unsigned jb = 1u << (j & 31);
    int jw = j >> 5;
#pragma unroll
    for (int r = 0; r < 8; r++) {
        int row = r + half * 8;  // C layout: lanes 0-15 -> M=r, lanes 16-31 -> M=r+8
        int i = i0 + row;
        float x1 = fmaxf(sbi[row * 4 + 0], bj0);
        float y1 = fmaxf(sbi[row * 4 + 1], bj1);
        float x2 = fminf(sbi[row * 4 + 2], bj2);
        float y2 = fminf(sbi[row * 4 + 3], bj3);
        float iw = fmaxf(x2 - x1, 0.0f), ih = fmaxf(y2 - y1, 0.0f);
        float inter = iw * ih;
        float denom = acc[r] - inter + 1e-9f;  // (area_i + area_j) - inter + eps
        bool pred = (inter > NMS_THR * denom) && (j > i) && (j < NDET) && (i < NDET);
        if (pred) atomicOr(&mask[i * WMSK + jw], jb);
    }
}

// ---------------- K7: sequential greedy scan + final outputs ----------------
__global__ void __launch_bounds__(256) k_scan_out(const float* __restrict__ s_score,
                                                  const int* __restrict__ s_label,
                                                  const int* __restrict__ s_valid,
                                                  const float* __restrict__ s_box,
                                                  const unsigned* __restrict__ mask,
                                                  float* __restrict__ out) {
    __shared__ unsigned supp[WMSK];
    __shared__ int alive;
    int tid = threadIdx.x;
    for (int w = tid; w < WMSK; w += blockDim.x) supp[w] = 0u;
    __syncthreads();
    for (int i = 0; i < NDET; i++) {
        if (tid == 0)
            alive = (s_valid[i] && !((supp[i >> 5] >> (i & 31)) & 1u)) ? 1 : 0;
        __syncthreads();
        if (alive) {
            for (int w = tid; w < WMSK; w += blockDim.x) supp[w] |= mask[i * WMSK + w];
        }
        __syncthreads();
    }
    float* ob = out;                 // boxes [3000*4]
    float* os = out + NDET * 4;      // scores [3000]
    float* ol = os + NDET;           // labels [3000]
    for (int i = tid; i < NDET; i += blockDim.x) {
        bool sup = (supp[i >> 5] >> (i & 31)) & 1u;
        bool keep = s_valid[i] && !sup;
#pragma unroll
        for (int k = 0; k < 4; k++) {
            float v = s_box[i * 4 + k] * (1.0f / IMGF);
            ob[i * 4 + k] = fminf(fmaxf(v, 0.0f), 1.0f);
        }
        os[i] = keep ? s_score[i] : 0.0f;
        ol[i] = keep ? (float)s_label[i] : -1.0f;
    }
}

// ---------------- host launcher ----------------
extern "C" void kernel_launch(void* const* d_in, const int* in_sizes, int n_in, void* d_out,
                              int out_size, void* d_ws, size_t ws_size, hipStream_t stream) {
    (void)n_in; (void)out_size; (void)ws_size;
    const float* obj[NLVL] = {(const float*)d_in[0], (const float*)d_in[4], (const float*)d_in[8]};
    const float* cls[NLVL] = {(const float*)d_in[1], (const float*)d_in[5], (const float*)d_in[9]};
    const float* reg[NLVL] = {(const float*)d_in[2], (const float*)d_in[6], (const float*)d_in[10]};
    const float* anc[NLVL] = {(const float*)d_in[3], (const float*)d_in[7], (const float*)d_in[11]};
    int M[NLVL] = {in_sizes[0], in_sizes[4], in_sizes[8]};  // obj has M elements
    const float strides[NLVL] = {8.0f, 16.0f, 32.0f};

    uint8_t* ws = (uint8_t*)d_ws;
    unsigned* hist = (unsigned*)(ws + OFF_HIST);
    int* meta = (int*)(ws + OFF_META);
    unsigned long long* cand = (unsigned long long*)(ws + OFF_CAND);
    float* det_score = (float*)(ws + OFF_DET_SCORE);
    int* det_label = (int*)(ws + OFF_DET_LABEL);
    int* det_valid = (int*)(ws + OFF_DET_VALID);
    float* det_box = (float*)(ws + OFF_DET_BOX);
    float* s_score = (float*)(ws + OFF_S_SCORE);
    int* s_label = (int*)(ws + OFF_S_LABEL);
    int* s_valid = (int*)(ws + OFF_S_VALID);
    float* s_box = (float*)(ws + OFF_S_BOX);
    float* s_bofs = (float*)(ws + OFF_S_BOFS);
    float* s_area = (float*)(ws + OFF_S_AREA);
    unsigned* mask = (unsigned*)(ws + OFF_MASK);

    int zwords = WS_END / 4;
    k_zero<<<512, 256, 0, stream>>>((unsigned*)ws, zwords);

    for (int l = 0; l < NLVL; l++) {
        int total = M[l] * NCLS;
        int blocks = (total + 255) / 256;
        if (blocks > 4096) blocks = 4096;
        k_score_hist<<<blocks, 256, 0, stream>>>(obj[l], cls[l], total, hist + l * NB);
    }
    k_thresh<<<1, 32, 0, stream>>>(hist, meta);
    for (int l = 0; l < NLVL; l++) {
        int total = M[l] * NCLS;
        int blocks = (total + 255) / 256;
        if (blocks > 4096) blocks = 4096;
        k_compact<<<blocks, 256, 0, stream>>>(obj[l], cls[l], total, l, meta, cand, reg[l], anc[l],
                                              strides[l], det_score, det_label, det_valid, det_box);
    }
    k_eq_select<<<NLVL, 256, 0, stream>>>(meta, cand, reg[0], reg[1], reg[2], anc[0], anc[1],
                                          anc[2], det_score, det_label, det_valid, det_box);
    k_sort_prep<<<1, 512, 0, stream>>>(det_score, det_label, det_valid, det_box, s_score, s_label,
                                       s_valid, s_box, s_bofs, s_area);
    dim3 tg(NPAD / 16, NPAD / 16);
    k_mask<<<tg, 32, 0, stream>>>(s_bofs, s_area, mask);
    k_scan_out<<<1, 256, 0, stream>>>(s_score, s_label, s_valid, s_box, mask, (float*)d_out);
}